// StereoTransformerNet_67834713473046
// MI455X (gfx1250) — compile-verified
//
#include <hip/hip_runtime.h>
#include <hip/hip_bf16.h>

typedef __attribute__((ext_vector_type(2))) float v2f;
typedef __attribute__((ext_vector_type(8))) float v8f;

#define BATCH 8
#define LSEQ  4096
#define DMODEL 384
#define NHEAD 8
#define NPTS  4
#define HDIM  48        // DMODEL / NHEAD
#define GRID  64        // sqrt(LSEQ)

#define BM 64
#define BN 64
#define BK 32
#define KDIM DMODEL     // all GEMMs have K = 384

// ---------------------------------------------------------------------------
// Tiled f32 WMMA GEMM:  Y[M,N] = A[M,KDIM] @ Bw[KDIM,N] + bias[N]
// Block: 256 threads (8 waves). Block tile 64x64, K-staged through LDS (BK=32).
// Each wave computes two 16x16 C tiles with V_WMMA_F32_16X16X4_F32.
// M multiple of 64, KDIM multiple of 32 (guaranteed); N guarded.
// ---------------------------------------------------------------------------
__global__ __launch_bounds__(256) void gemm_wmma_f32(
    const float* __restrict__ A, const float* __restrict__ Bw,
    const float* __restrict__ bias, float* __restrict__ Y, int N)
{
    __shared__ float As[BM][BK + 2];   // pad 2 -> 8B-aligned frag reads, no bank conflicts
    __shared__ float Bs[BK][BN];

    const int tid   = threadIdx.x;
    const int m0    = blockIdx.x * BM;
    const int n0    = blockIdx.y * BN;
    const int w     = tid >> 5;        // wave 0..7
    const int lane  = tid & 31;
    const int lrow  = lane & 15;       // M (A/C) or N (B) index within tile
    const int lhalf = lane >> 4;       // 0: K=0..1 / M=0..7 ; 1: K=2..3 / M=8..15
    const int mi    = w & 3;           // subtile row 0..3
    const int ni0   = (w >> 2) * 2;    // subtile cols {0,1} or {2,3}
    const int ni1   = ni0 + 1;

    v8f c0 = {};
    v8f c1 = {};

    for (int k0 = 0; k0 < KDIM; k0 += BK) {
        // ---- stage A tile (64x32): 512 float4, 2 per thread ----
        #pragma unroll
        for (int rep = 0; rep < 2; ++rep) {
            int idx = tid + rep * 256;
            int r   = idx >> 3;            // 0..63
            int c4  = (idx & 7) << 2;      // 0..28
            const float4 a4 = *(const float4*)(A + (size_t)(m0 + r) * KDIM + k0 + c4);
            As[r][c4 + 0] = a4.x; As[r][c4 + 1] = a4.y;
            As[r][c4 + 2] = a4.z; As[r][c4 + 3] = a4.w;
        }
        // ---- stage B tile (32x64): 512 float4, 2 per thread (N-guarded) ----
        #pragma unroll
        for (int rep = 0; rep < 2; ++rep) {
            int idx = tid + rep * 256;
            int r   = idx >> 4;            // 0..31
            int c4  = (idx & 15) << 2;     // 0..60
            float4 b4 = make_float4(0.f, 0.f, 0.f, 0.f);
            if (n0 + c4 < N)
                b4 = *(const float4*)(Bw + (size_t)(k0 + r) * N + n0 + c4);
            Bs[r][c4 + 0] = b4.x; Bs[r][c4 + 1] = b4.y;
            Bs[r][c4 + 2] = b4.z; Bs[r][c4 + 3] = b4.w;
        }
        __syncthreads();

        // prefetch next K tile into cache (global_prefetch_b8)
        if (k0 + BK < KDIM) {
            __builtin_prefetch(A + (size_t)(m0 + (tid >> 3)) * KDIM + (k0 + BK) + ((tid & 7) << 2), 0, 1);
            if (n0 + ((tid & 15) << 2) < N)
                __builtin_prefetch(Bw + (size_t)(k0 + BK + (tid >> 4)) * N + n0 + ((tid & 15) << 2), 0, 1);
        }

        // ---- 8 x V_WMMA_F32_16X16X4_F32 steps over this K stage ----
        #pragma unroll
        for (int kt = 0; kt < BK / 4; ++kt) {
            const int kk = kt * 4 + 2 * lhalf;
            // A frag (16x4): lane<16 -> K=0,1 ; lane>=16 -> K=2,3
            v2f afr;
            afr[0] = As[mi * 16 + lrow][kk];
            afr[1] = As[mi * 16 + lrow][kk + 1];
            // B frags (4x16): lane<16 -> K=0,1 rows ; lane>=16 -> K=2,3 rows
            v2f bf0;
            bf0[0] = Bs[kk][ni0 * 16 + lrow];
            bf0[1] = Bs[kk + 1][ni0 * 16 + lrow];
            c0 = __builtin_amdgcn_wmma_f32_16x16x4_f32(false, afr, false, bf0,
                                                       (short)0, c0, false, false);
            v2f bf1;
            bf1[0] = Bs[kk][ni1 * 16 + lrow];
            bf1[1] = Bs[kk + 1][ni1 * 16 + lrow];
            c1 = __builtin_amdgcn_wmma_f32_16x16x4_f32(false, afr, false, bf1,
                                                       (short)0, c1, false, false);
        }
        __syncthreads();
    }

    // ---- store C (+bias): VGPR r -> row (r + 8*lhalf), col lrow ----
    #pragma unroll
    for (int r = 0; r < 8; ++r) {
        const int gm  = m0 + mi * 16 + r + 8 * lhalf;
        const int gn0 = n0 + ni0 * 16 + lrow;
        const int gn1 = n0 + ni1 * 16 + lrow;
        if (gn0 < N) Y[(size_t)gm * N + gn0] = c0[r] + bias[gn0];
        if (gn1 < N) Y[(size_t)gm * N + gn1] = c1[r] + bias[gn1];
    }
}

// ---------------------------------------------------------------------------
// Deformable bilinear sampling + softmax point weights + aggregation.
// One block per (b,l); 32 threads precompute per-(head,point) params into LDS,
// then 384 threads produce one output channel each:
//   mid[b,l,h*48+d] = sum_p wt[h,p] * bilinear(v[b,:,h*48+d], loc[h,p])
// ---------------------------------------------------------------------------
__global__ __launch_bounds__(384) void deform_sample(
    const float* __restrict__ v, const float* __restrict__ offr,
    const float* __restrict__ wtr, float* __restrict__ mid)
{
    __shared__ float s_wx[32], s_wy[32], s_wt[32], s_logit[32];
    __shared__ int   s_t00[32], s_t01[32], s_t10[32], s_t11[32];

    const int tid = threadIdx.x;
    const int bl  = blockIdx.x;           // b*LSEQ + l
    const int l   = bl & (LSEQ - 1);
    const int blB = bl - l;               // b*LSEQ

    if (tid < 32) {                       // one thread per (head, point)
        const float offx = offr[(size_t)bl * (NHEAD * NPTS * 2) + (tid << 1) + 0];
        const float offy = offr[(size_t)bl * (NHEAD * NPTS * 2) + (tid << 1) + 1];
        s_logit[tid] = wtr[(size_t)bl * (NHEAD * NPTS) + tid];

        const int   gi = l >> 6, gj = l & (GRID - 1);
        const float inv = 1.0f / (float)(GRID - 1);
        const float locx = fminf(fmaxf(gj * inv + offx, 0.f), 1.f);  // component 0
        const float locy = fminf(fmaxf(gi * inv + offy, 0.f), 1.f);  // component 1
        // faithful to source: component 0 scaled by (h-1) and used as the ROW coord
        const float ph = locx * (float)(GRID - 1);
        const float pw = locy * (float)(GRID - 1);
        const float y0 = floorf(ph), x0 = floorf(pw);
        const int iy0 = (int)y0, ix0 = (int)x0;
        const int iy1 = min(iy0 + 1, GRID - 1);
        const int ix1 = min(ix0 + 1, GRID - 1);
        s_wy[tid]  = ph - y0;
        s_wx[tid]  = pw - x0;
        s_t00[tid] = iy0 * GRID + ix0;
        s_t01[tid] = iy0 * GRID + ix1;
        s_t10[tid] = iy1 * GRID + ix0;
        s_t11[tid] = iy1 * GRID + ix1;
    }
    __syncthreads();
    if (tid < 32) {                       // softmax over the 4 points of my head
        const int h4 = tid & ~3;
        const float l0 = s_logit[h4 + 0], l1 = s_logit[h4 + 1];
        const float l2 = s_logit[h4 + 2], l3 = s_logit[h4 + 3];
        const float mx = fmaxf(fmaxf(l0, l1), fmaxf(l2, l3));
        const float den = expf(l0 - mx) + expf(l1 - mx) + expf(l2 - mx) + expf(l3 - mx);
        s_wt[tid] = expf(s_logit[tid] - mx) / den;
    }
    __syncthreads();

    const int c  = tid;                   // output channel == h*HDIM + d
    const int h_ = c / HDIM;
    float acc = 0.f;
    #pragma unroll
    for (int p = 0; p < NPTS; ++p) {
        const int hp = (h_ << 2) + p;
        const float wx = s_wx[hp], wy = s_wy[hp], wgt = s_wt[hp];
        const float v00 = v[((size_t)(blB + s_t00[hp])) * DMODEL + c];
        const float v01 = v[((size_t)(blB + s_t01[hp])) * DMODEL + c];
        const float v10 = v[((size_t)(blB + s_t10[hp])) * DMODEL + c];
        const float v11 = v[((size_t)(blB + s_t11[hp])) * DMODEL + c];
        const float top = (1.f - wx) * v00 + wx * v01;
        const float bot = (1.f - wx) * v10 + wx * v11;
        acc += wgt * ((1.f - wy) * top + wy * bot);
    }
    mid[(size_t)bl * DMODEL + c] = acc;
}

// ---------------------------------------------------------------------------
extern "C" void kernel_launch(void* const* d_in, const int* in_sizes, int n_in,
                              void* d_out, int out_size, void* d_ws, size_t ws_size,
                              hipStream_t stream) {
    const float* query = (const float*)d_in[0];
    // d_in[1] (key), d_in[3..6] (Wq,bq,Wk,bk) are dead in the reference output
    const float* value = (const float*)d_in[2];
    const float* Wv    = (const float*)d_in[7];
    const float* bv    = (const float*)d_in[8];
    const float* Woff  = (const float*)d_in[9];
    const float* boff  = (const float*)d_in[10];
    const float* Wwt   = (const float*)d_in[11];
    const float* bwt   = (const float*)d_in[12];
    const float* Wo    = (const float*)d_in[13];
    const float* bo    = (const float*)d_in[14];
    float* out = (float*)d_out;

    const size_t M = (size_t)BATCH * LSEQ;       // 32768 rows
    char* ws = (char*)d_ws;
    float* vproj = (float*)(ws);                                             // M*384
    float* offr  = (float*)(ws + M * DMODEL * 4);                            // M*64
    float* wtr   = (float*)(ws + M * DMODEL * 4 + M * 64 * 4);               // M*32
    float* mid   = (float*)(ws + M * DMODEL * 4 + M * 64 * 4 + M * 32 * 4);  // M*384

    const dim3 blk(256);
    // v = value @ Wv + bv
    gemm_wmma_f32<<<dim3(M / BM, DMODEL / BN), blk, 0, stream>>>(value, Wv, bv, vproj, DMODEL);
    // off logits = query @ Woff + boff   (N = 64)
    gemm_wmma_f32<<<dim3(M / BM, 1), blk, 0, stream>>>(query, Woff, boff, offr, NHEAD * NPTS * 2);
    // weight logits = query @ Wwt + bwt  (N = 32)
    gemm_wmma_f32<<<dim3(M / BM, 1), blk, 0, stream>>>(query, Wwt, bwt, wtr, NHEAD * NPTS);
    // deformable sampling + softmax + aggregation -> mid
    deform_sample<<<dim3((unsigned)M), dim3(DMODEL), 0, stream>>>(vproj, offr, wtr, mid);
    // out = mid @ Wo + bo
    gemm_wmma_f32<<<dim3(M / BM, DMODEL / BN), blk, 0, stream>>>(mid, Wo, bo, out, DMODEL);
}